// Voxelization_867583394203
// MI455X (gfx1250) — compile-verified
//
#include <hip/hip_runtime.h>
#include <hip/hip_bf16.h>
#include <stdint.h>

// Voxelization scatter-mean for MI455X (gfx1250).
// - Full 32^3 fp32 accumulator (128KB) per workgroup in CDNA5's 320KB LDS:
//   all scatter atomics are ds_add_f32 (LDS pipeline), zero global atomics.
// - Feature stream staged through LDS with double-buffered
//   global_load_async_to_lds_b128 (ASYNCcnt) to overlap HBM latency with
//   the LDS-atomic work.
// - All bulk traffic as 128-bit ops.

#define RES   32
#define RES3  (RES * RES * RES)   // 32768
#define BB    8
#define CC    64

#if defined(__gfx1250__) && defined(__HIP_DEVICE_COMPILE__) && \
    __has_builtin(__builtin_amdgcn_s_wait_asynccnt)
  #define AS_GLOBAL __attribute__((address_space(1)))
  #define AS_LDS    __attribute__((address_space(3)))
  typedef int v4i __attribute__((__vector_size__(4 * sizeof(int))));
  #if __has_builtin(__builtin_amdgcn_global_load_async_to_lds_b128)
    #define ASYNC_B128_OK 1
  #else
    #define ASYNC_B128_OK 0
  #endif
  #if __has_builtin(__builtin_amdgcn_global_load_async_to_lds_b32)
    #define ASYNC_B32_OK 1
  #else
    #define ASYNC_B32_OK 0
  #endif
#else
  #define ASYNC_B128_OK 0
  #define ASYNC_B32_OK 0
#endif

// ---------------------------------------------------------------------------
// Kernel 1: per-(batch, axis) mean of coords over N (24 reductions).
// float4 main loop + scalar tail, LDS tree reduce.
// ---------------------------------------------------------------------------
__global__ void __launch_bounds__(256)
k_mean(const float* __restrict__ coords, float* __restrict__ means, int N) {
    __shared__ float red[256];
    const int ba = blockIdx.x;                      // 0..23 = b*3 + axis
    const float* p = coords + (size_t)ba * N;
    const int N4 = N >> 2;
    const float4* p4 = (const float4*)p;
    float s = 0.0f;
    for (int n = threadIdx.x; n < N4; n += blockDim.x) {
        float4 v = p4[n];
        s += (v.x + v.y) + (v.z + v.w);
    }
    for (int n = (N4 << 2) + threadIdx.x; n < N; n += blockDim.x) s += p[n];
    red[threadIdx.x] = s;
    __syncthreads();
    for (int off = 128; off > 0; off >>= 1) {
        if ((int)threadIdx.x < off) red[threadIdx.x] += red[threadIdx.x + off];
        __syncthreads();
    }
    if (threadIdx.x == 0) means[ba] = red[0] / (float)N;
}

// ---------------------------------------------------------------------------
// Kernel 2: norm_coords (second output) + flat voxel index per point.
// jnp.round == round-half-to-even == rintf (default RNE mode).
// ---------------------------------------------------------------------------
__global__ void __launch_bounds__(256)
k_norm_idx(const float* __restrict__ coords, const float* __restrict__ means,
           float* __restrict__ norm_out, int* __restrict__ flat_idx, int N) {
    const int t = blockIdx.x * blockDim.x + threadIdx.x;
    if (t >= BB * N) return;
    const int b = t / N;
    const int n = t - b * N;
    int vi[3];
#pragma unroll
    for (int a = 0; a < 3; ++a) {
        const size_t off = ((size_t)b * 3 + a) * (size_t)N + n;
        float nc = (coords[off] - means[b * 3 + a] + 1.0f) * 0.5f * (float)RES;
        nc = fminf(fmaxf(nc, 0.0f), (float)(RES - 1));
        norm_out[off] = nc;
        vi[a] = (int)rintf(nc);
    }
    flat_idx[t] = vi[0] * (RES * RES) + vi[1] * RES + vi[2];
}

// ---------------------------------------------------------------------------
// Kernel 3: per-batch voxel occupancy counts. One 1024-thread workgroup per
// batch owns the full 32^3 counter grid in LDS (128KB); int4 index loads;
// flush is a plain store (zero global atomics).
// ---------------------------------------------------------------------------
__global__ void __launch_bounds__(1024)
k_count(const int* __restrict__ flat_idx, float* __restrict__ counts, int N) {
    __shared__ float cnt[RES3];                     // 128 KB
    const int b = blockIdx.x;
    for (int v = threadIdx.x; v < RES3; v += blockDim.x) cnt[v] = 0.0f;
    __syncthreads();
    const int* idx = flat_idx + (size_t)b * N;
    const int N4 = N >> 2;
    const int4* idx4 = (const int4*)idx;
    for (int n = threadIdx.x; n < N4; n += blockDim.x) {
        int4 i4 = idx4[n];
        atomicAdd(&cnt[i4.x], 1.0f);
        atomicAdd(&cnt[i4.y], 1.0f);
        atomicAdd(&cnt[i4.z], 1.0f);
        atomicAdd(&cnt[i4.w], 1.0f);
    }
    for (int n = (N4 << 2) + threadIdx.x; n < N; n += blockDim.x)
        atomicAdd(&cnt[idx[n]], 1.0f);
    __syncthreads();
    float* out = counts + (size_t)b * RES3;
    for (int v = threadIdx.x; v < RES3; v += blockDim.x) out[v] = cnt[v];
}

// ---------------------------------------------------------------------------
// Kernel 4: scatter-sum one (batch, channel) into a 128KB LDS accumulator.
// Features stream through a double-buffered async-to-LDS staging buffer:
// issue chunk i+1 (b128/lane), s_wait_asynccnt(1) -> chunk i resident,
// process chunk i with ds_add_f32 while chunk i+1 is in flight.
// Then vox = sum / max(count,1) written straight to d_out (full coverage).
// ---------------------------------------------------------------------------
#define SCATTER_T 256
#define CHUNK     (SCATTER_T * 4)                   // 1024 points / chunk

__global__ void __launch_bounds__(SCATTER_T)
k_scatter(const float* __restrict__ feats, const int* __restrict__ flat_idx,
          const float* __restrict__ counts, float* __restrict__ vox, int N) {
    __shared__ float acc[RES3];                     // 128 KB accumulator
#if ASYNC_B128_OK || ASYNC_B32_OK
    __shared__ __align__(16) float stage[2][CHUNK]; // 8 KB staging
#endif
    const int c   = blockIdx.x;
    const int b   = blockIdx.y;
    const int tid = threadIdx.x;

    for (int v = tid; v < RES3; v += SCATTER_T) acc[v] = 0.0f;
    __syncthreads();

    const float* f   = feats + ((size_t)b * CC + c) * (size_t)N;
    const int*   idx = flat_idx + (size_t)b * N;
    const int nFull  = N / CHUNK;                   // full 1024-point chunks

#if ASYNC_B128_OK
    if (nFull > 0) {
        __builtin_amdgcn_global_load_async_to_lds_b128(
            (AS_GLOBAL v4i*)(f + (size_t)tid * 4),
            (AS_LDS v4i*)&stage[0][tid * 4], 0, 0);
    }
    for (int ch = 0; ch < nFull; ++ch) {
        const int cur = ch & 1;
        if (ch + 1 < nFull) {
            __builtin_amdgcn_global_load_async_to_lds_b128(
                (AS_GLOBAL v4i*)(f + (size_t)(ch + 1) * CHUNK + tid * 4),
                (AS_LDS v4i*)&stage[cur ^ 1][tid * 4], 0, 0);
            __builtin_amdgcn_s_wait_asynccnt(1);    // chunk ch resident
        } else {
            __builtin_amdgcn_s_wait_asynccnt(0);
        }
        asm volatile("" ::: "memory");              // keep LDS reads below wait
        const float4 v4 = *(const float4*)&stage[cur][tid * 4];
        const int4   i4 = *(const int4*)&idx[ch * CHUNK + tid * 4];
        atomicAdd(&acc[i4.x], v4.x);                // ds_add_f32 (no rtn)
        atomicAdd(&acc[i4.y], v4.y);
        atomicAdd(&acc[i4.z], v4.z);
        atomicAdd(&acc[i4.w], v4.w);
    }
#else
    for (int ch = 0; ch < nFull; ++ch) {
        const int n0 = ch * CHUNK + tid * 4;
        float4 v4;
#if ASYNC_B32_OK
        __builtin_amdgcn_global_load_async_to_lds_b32(
            (AS_GLOBAL int*)(f + n0), (AS_LDS int*)&stage[0][tid * 4], 0, 0);
        __builtin_amdgcn_global_load_async_to_lds_b32(
            (AS_GLOBAL int*)(f + n0 + 1), (AS_LDS int*)&stage[0][tid * 4 + 1], 0, 0);
        __builtin_amdgcn_global_load_async_to_lds_b32(
            (AS_GLOBAL int*)(f + n0 + 2), (AS_LDS int*)&stage[0][tid * 4 + 2], 0, 0);
        __builtin_amdgcn_global_load_async_to_lds_b32(
            (AS_GLOBAL int*)(f + n0 + 3), (AS_LDS int*)&stage[0][tid * 4 + 3], 0, 0);
        __builtin_amdgcn_s_wait_asynccnt(0);
        asm volatile("" ::: "memory");
        v4 = *(const float4*)&stage[0][tid * 4];
#else
        v4 = *(const float4*)&f[n0];
#endif
        const int4 i4 = *(const int4*)&idx[n0];
        atomicAdd(&acc[i4.x], v4.x);
        atomicAdd(&acc[i4.y], v4.y);
        atomicAdd(&acc[i4.z], v4.z);
        atomicAdd(&acc[i4.w], v4.w);
    }
#endif
    // Scalar tail (N % 1024 points), plain loads.
    for (int n = nFull * CHUNK + tid; n < N; n += SCATTER_T)
        atomicAdd(&acc[idx[n]], f[n]);

    __syncthreads();

    const float* cb = counts + (size_t)b * RES3;
    float* ob = vox + ((size_t)b * CC + c) * (size_t)RES3;
    for (int v = tid; v < RES3; v += SCATTER_T)
        ob[v] = acc[v] / fmaxf(cb[v], 1.0f);
}

// ---------------------------------------------------------------------------
// Host launcher.
// d_in[0] = features [8,64,N] f32, d_in[1] = coords [8,3,N] f32.
// d_out   = vox [8,64,32768] f32  ++  norm_coords [8,3,N] f32.
// ws      : [0..32) means | [32..32+8*32768) counts | then flat_idx (int).
// ---------------------------------------------------------------------------
extern "C" void kernel_launch(void* const* d_in, const int* in_sizes, int n_in,
                              void* d_out, int out_size, void* d_ws, size_t ws_size,
                              hipStream_t stream) {
    (void)n_in; (void)out_size; (void)ws_size;

    const float* feats  = (const float*)d_in[0];
    const float* coords = (const float*)d_in[1];
    const int N = in_sizes[0] / (BB * CC);          // 100000

    float* vox      = (float*)d_out;                            // 8*64*32768
    float* norm_out = (float*)d_out + (size_t)BB * CC * RES3;   // 8*3*N

    float* ws       = (float*)d_ws;
    float* means    = ws;                                       // 24 (pad 32)
    float* counts   = ws + 32;                                  // 8*32768
    int*   flat_idx = (int*)(ws + 32 + (size_t)BB * RES3);      // 8*N

    k_mean<<<BB * 3, 256, 0, stream>>>(coords, means, N);

    const int total = BB * N;
    k_norm_idx<<<(total + 255) / 256, 256, 0, stream>>>(coords, means,
                                                        norm_out, flat_idx, N);

    k_count<<<BB, 1024, 0, stream>>>(flat_idx, counts, N);

    k_scatter<<<dim3(CC, BB), SCATTER_T, 0, stream>>>(feats, flat_idx, counts,
                                                      vox, N);
}